// BA_28784870818378
// MI455X (gfx1250) — compile-verified
//
#include <hip/hip_runtime.h>
#include <hip/hip_bf16.h>

typedef __attribute__((ext_vector_type(2))) float v2f;
typedef __attribute__((ext_vector_type(8))) float v8f;

__device__ __forceinline__ v8f wmma_f32(v2f a, v2f b, v8f c) {
  // D = A(16x4) * B(4x16) + C, exact fp32
  return __builtin_amdgcn_wmma_f32_16x16x4_f32(false, a, false, b, (short)0, c, false, false);
}

// ---------------- Kernel 1: window mean of x -> xmean (2,1024,192) ----------------
__global__ void win_mean_kernel(const float* __restrict__ x, float* __restrict__ xmean) {
  int blk = blockIdx.x;               // b*1024 + n
  int b = blk >> 10, n = blk & 1023;
  int nh = n >> 5, nw = n & 31;
  const float* xb = x + (size_t)b * 192 * 65536;
  __shared__ float s[192];
  int t = threadIdx.x;
  if (t < 192) s[t] = 0.f;
  __syncthreads();
  int base = (nh * 8) * 256 + nw * 8;
  #pragma unroll
  for (int i = 0; i < 6; ++i) {       // 1536 (c,row) pairs / 256 threads
    int id = i * 256 + t;
    int c = id >> 3, row = id & 7;
    const float* p = xb + (size_t)c * 65536 + base + row * 256;
    float s8 = 0.f;
    #pragma unroll
    for (int j = 0; j < 8; ++j) s8 += p[j];
    atomicAdd(&s[c], s8);
  }
  __syncthreads();
  if (t < 192) xmean[((size_t)b * 1024 + n) * 192 + t] = s[t] * (1.0f / 64.0f);
}

// ------- Kernel 2: qkwin = xmean @ w_qkv[:, :384] + bias, q part pre-scaled -------
// grid 128 (b*64+mtile), block 128 (4 waves, each 96 N-cols)
__global__ void qkwin_kernel(const float* __restrict__ xmean, const float* __restrict__ w_qkv,
                             const float* __restrict__ b_qkv, float* __restrict__ qkwin) {
  int blk = blockIdx.x; int b = blk >> 6; int mt = blk & 63;
  int lane = threadIdx.x & 31; int wave = threadIdx.x >> 5;
  int m0 = mt * 16, n0 = wave * 96;
  const float* A = xmean + ((size_t)b * 1024 + m0) * 192;
  v8f acc[6] = {};
  int am = lane & 15, kk = (lane >> 4) * 2;
  for (int k0 = 0; k0 < 192; k0 += 4) {
    v2f a;
    a.x = A[am * 192 + k0 + kk];
    a.y = A[am * 192 + k0 + kk + 1];
    #pragma unroll
    for (int nt = 0; nt < 6; ++nt) {
      int col = n0 + nt * 16 + am;
      v2f bb;
      bb.x = w_qkv[(size_t)(k0 + kk) * 576 + col];
      bb.y = w_qkv[(size_t)(k0 + kk + 1) * 576 + col];
      acc[nt] = wmma_f32(a, bb, acc[nt]);
    }
  }
  const float scale = 0.0721687836487032f;  // 192^-0.5
  #pragma unroll
  for (int nt = 0; nt < 6; ++nt) {
    int col = n0 + nt * 16 + am;
    float bias = b_qkv[col];
    float sc = (col < 192) ? scale : 1.0f;
    int rbase = 8 * (lane >> 4);
    #pragma unroll
    for (int i = 0; i < 8; ++i) {
      int row = m0 + rbase + i;
      qkwin[((size_t)b * 1024 + row) * 384 + col] = (acc[nt][i] + bias) * sc;
    }
  }
}

// ---------- Kernel 3: logits[b] = qwin_scaled @ kwin^T  (1024x1024, K=192) ----------
// grid 128 (b*64+mtile), block 256 (8 waves, each 128 N-cols)
__global__ void logits_kernel(const float* __restrict__ qkwin, float* __restrict__ logits) {
  int blk = blockIdx.x; int b = blk >> 6; int mt = blk & 63;
  int lane = threadIdx.x & 31; int wave = threadIdx.x >> 5;
  int m0 = mt * 16, n0 = wave * 128;
  const float* Q = qkwin + ((size_t)b * 1024 + m0) * 384;        // q part: cols 0..191
  const float* K = qkwin + ((size_t)b * 1024) * 384 + 192;       // k part: cols 192..383
  v8f acc[8] = {};
  int am = lane & 15, kk = (lane >> 4) * 2;
  for (int k0 = 0; k0 < 192; k0 += 4) {
    v2f a;
    a.x = Q[am * 384 + k0 + kk];
    a.y = Q[am * 384 + k0 + kk + 1];
    #pragma unroll
    for (int nt = 0; nt < 8; ++nt) {
      int col = n0 + nt * 16 + am;                               // key window index
      v2f bb;
      bb.x = K[(size_t)col * 384 + k0 + kk];
      bb.y = K[(size_t)col * 384 + k0 + kk + 1];
      acc[nt] = wmma_f32(a, bb, acc[nt]);
    }
  }
  #pragma unroll
  for (int nt = 0; nt < 8; ++nt) {
    int col = n0 + nt * 16 + am;
    int rbase = 8 * (lane >> 4);
    #pragma unroll
    for (int i = 0; i < 8; ++i) {
      int row = m0 + rbase + i;
      logits[((size_t)b * 1024 + row) * 1024 + col] = acc[nt][i];
    }
  }
}

// ---------------- Kernel 4: top-4 per logits row (wave per row) ----------------
// grid 256, block 256 (8 waves); jax tie-break: equal values -> smaller index
__global__ void topk_kernel(const float* __restrict__ logits, int* __restrict__ topk) {
  int lane = threadIdx.x & 31; int wave = threadIdx.x >> 5;
  int row = blockIdx.x * 8 + wave;            // 0..2047 == b*1024+n
  const float* L = logits + (size_t)row * 1024;
  float vals[32];
  #pragma unroll
  for (int i = 0; i < 32; ++i) vals[i] = L[i * 32 + lane];
  unsigned used = 0u;
  for (int j = 0; j < 4; ++j) {
    float bv = -__builtin_inff(); int bi = 0x7fffffff;
    #pragma unroll
    for (int i = 0; i < 32; ++i) {
      if (!(used & (1u << i))) {
        float v = vals[i]; int idx = i * 32 + lane;
        if (v > bv || (v == bv && idx < bi)) { bv = v; bi = idx; }
      }
    }
    #pragma unroll
    for (int off = 16; off > 0; off >>= 1) {
      float ov = __shfl_xor(bv, off, 32);
      int   oi = __shfl_xor(bi, off, 32);
      if (ov > bv || (ov == bv && oi < bi)) { bv = ov; bi = oi; }
    }
    if ((bi & 31) == lane) used |= (1u << (bi >> 5));
    if (lane == 0) topk[row * 4 + j] = bi;
  }
}

// -------- Kernel 5: fused v-projection + gather: out[b,n,j] = win(x)[idx] @ w_v + b_v --------
// grid 8192 (b*4096 + n*4 + j), block 128 (4 waves, each 16 pixels x 192 out-cols)
__global__ void vgather_kernel(const float* __restrict__ x, const float* __restrict__ w_qkv,
                               const float* __restrict__ b_qkv, const int* __restrict__ topk,
                               float* __restrict__ out) {
  int blk = blockIdx.x;
  int b = blk >> 12; int rem = blk & 4095; int n = rem >> 2; int j = rem & 3;
  int lane = threadIdx.x & 31; int wave = threadIdx.x >> 5;
  int m = topk[(((size_t)b * 1024) + n) * 4 + j];      // selected source window
  int mh = m >> 5, mw = m & 31;
  const float* xb = x + (size_t)b * 192 * 65536;
  int p0 = wave * 16;
  int am = lane & 15, kk = (lane >> 4) * 2;
  int p = p0 + am;
  int off = (mh * 8 + (p >> 3)) * 256 + mw * 8 + (p & 7);
  v8f acc[12] = {};
  for (int k0 = 0; k0 < 192; k0 += 4) {
    v2f a;
    a.x = xb[(size_t)(k0 + kk) * 65536 + off];
    a.y = xb[(size_t)(k0 + kk + 1) * 65536 + off];
    #pragma unroll
    for (int nt = 0; nt < 12; ++nt) {
      int col = 384 + nt * 16 + am;                    // v columns of w_qkv
      v2f bb;
      bb.x = w_qkv[(size_t)(k0 + kk) * 576 + col];
      bb.y = w_qkv[(size_t)(k0 + kk + 1) * 576 + col];
      acc[nt] = wmma_f32(a, bb, acc[nt]);
    }
  }
  float* O = out + (((size_t)b * 1024 + n) * 4 + j) * 64 * 192;
  #pragma unroll
  for (int nt = 0; nt < 12; ++nt) {
    int d = nt * 16 + am;
    float bias = b_qkv[384 + d];
    int rbase = 8 * (lane >> 4);
    #pragma unroll
    for (int i = 0; i < 8; ++i) {
      int pix = p0 + rbase + i;
      O[(size_t)pix * 192 + d] = acc[nt][i] + bias;   // 64B-contiguous per half-wave
    }
  }
}

extern "C" void kernel_launch(void* const* d_in, const int* in_sizes, int n_in,
                              void* d_out, int out_size, void* d_ws, size_t ws_size,
                              hipStream_t stream) {
  const float* x     = (const float*)d_in[0];   // (2,192,256,256)
  const float* w_qkv = (const float*)d_in[1];   // (192,576)
  const float* b_qkv = (const float*)d_in[2];   // (576,)
  float* out = (float*)d_out;                   // (2,1024,4,64,192)

  float* xmean  = (float*)d_ws;                 // 2*1024*192
  float* qkwin  = xmean + (size_t)2 * 1024 * 192;     // 2*1024*384
  float* logits = qkwin + (size_t)2 * 1024 * 384;     // 2*1024*1024
  int*   tk     = (int*)(logits + (size_t)2 * 1024 * 1024); // 2*1024*4

  win_mean_kernel<<<2048, 256, 0, stream>>>(x, xmean);
  qkwin_kernel  <<< 128, 128, 0, stream>>>(xmean, w_qkv, b_qkv, qkwin);
  logits_kernel <<< 128, 256, 0, stream>>>(qkwin, logits);
  topk_kernel   <<< 256, 256, 0, stream>>>(logits, tk);
  vgather_kernel<<<8192, 128, 0, stream>>>(x, w_qkv, b_qkv, tk, out);
}